// RHF_35708358099211
// MI455X (gfx1250) — compile-verified
//
#include <hip/hip_runtime.h>

#define NBF   96
#define NBF2  (NBF * NBF)     // 9216
#define NITER 20
#define SQ_ITERS 12           // B^(2^12): power iteration via WMMA matrix squaring

typedef __attribute__((ext_vector_type(2))) float v2f;
typedef __attribute__((ext_vector_type(8))) float v8f;

#if __has_builtin(__builtin_amdgcn_global_load_async_to_lds_b128)
#define HAVE_ASYNC_LDS 1
typedef int v4i_vs __attribute__((vector_size(16)));
typedef __attribute__((address_space(1))) v4i_vs* gvec_t;
typedef __attribute__((address_space(3))) v4i_vs* lvec_t;
#else
#define HAVE_ASYNC_LDS 0
#endif

__device__ __forceinline__ void wait_async0() {
#if __has_builtin(__builtin_amdgcn_s_wait_asynccnt)
  __builtin_amdgcn_s_wait_asynccnt(0);
#else
  asm volatile("s_wait_asynccnt 0x0" ::: "memory");
#endif
}

// ---------- helpers ----------
__device__ __forceinline__ float reduce96(float* red, int t, float val) {
  red[t] = val;                       __syncthreads();
  if (t < 32) red[t] += red[t + 32] + red[t + 64];   __syncthreads();
  if (t < 8)  red[t] += red[t + 8] + red[t + 16] + red[t + 24]; __syncthreads();
  if (t < 2)  red[t] += red[t + 2] + red[t + 4] + red[t + 6];   __syncthreads();
  if (t == 0) red[0] += red[1];       __syncthreads();
  float r = red[0];                   __syncthreads();
  return r;
}

__global__ __launch_bounds__(256) void zero_kernel(float* __restrict__ p, int n) {
  int i = blockIdx.x * 256 + threadIdx.x;
  if (i < n) p[i] = 0.0f;
}

// ---------- fused J/K contraction: single pass over G (HBM-bound) ----------
// Block (p,a) stages G[p,a,:,:] (36 KB) in padded LDS via async global->LDS b128.
//   J[p,a]  = sum_{k,l} G[p,a,k,l] * D[k,l]          (block-owned, one store)
//   K[p,k] += sum_l     G[p,a,k,l] * D[a,l]          (one float atomic per thread)
__global__ __launch_bounds__(NBF) void jk_kernel(const float* __restrict__ G,
                                                 const float* __restrict__ D,
                                                 float* __restrict__ J,
                                                 float* __restrict__ K) {
  __shared__ float g[NBF * 97];   // pad row stride to 97 -> conflict-free
  __shared__ float da[NBF];
  __shared__ float red[NBF];
  const int t = threadIdx.x;
  const int p = blockIdx.x / NBF;
  const int a = blockIdx.x - p * NBF;
  const float4* Gpa = (const float4*)(G + (size_t)(p * NBF + a) * NBF2);

#if HAVE_ASYNC_LDS
  // Async tensor-path staging: one b128 per lane per step, no VGPR round-trip.
  for (int i4 = t; i4 < NBF2 / 4; i4 += NBF) {
    int i = i4 << 2;
    int r = i / NBF;
    int s = i - r * NBF;                          // 96 % 4 == 0: chunk stays in-row
    __builtin_amdgcn_global_load_async_to_lds_b128(
        (gvec_t)(Gpa + i4), (lvec_t)(g + r * 97 + s), 0, 0);
  }
  da[t] = D[a * NBF + t];
  wait_async0();                                  // drain ASYNCcnt -> LDS valid
  __syncthreads();
#else
  for (int i4 = t; i4 < NBF2 / 4; i4 += NBF) {   // coalesced b128 loads
    float4 gv = Gpa[i4];
    int i = i4 << 2;
    int r = i / NBF;
    int s = i - r * NBF;
    float* gd = g + r * 97 + s;
    gd[0] = gv.x; gd[1] = gv.y; gd[2] = gv.z; gd[3] = gv.w;
  }
  da[t] = D[a * NBF + t];
  __syncthreads();
#endif

  const float* Dr = D + t * NBF;                  // hot 36 KB, lives in WGP$/L2
  const float* gr = g + t * 97;
  float jp = 0.0f, kp = 0.0f;
  #pragma unroll 4
  for (int l = 0; l < NBF; ++l) {
    float gv = gr[l];
    jp = fmaf(gv, Dr[l], jp);
    kp = fmaf(gv, da[l], kp);
  }
  float jsum = reduce96(red, t, jp);
  if (t == 0) J[p * NBF + a] = jsum;
  atomicAdd(&K[p * NBF + t], kp);
}

// ---------- F = H + 2J - K ----------
__global__ __launch_bounds__(256) void fock_kernel(const float* __restrict__ H,
                                                   const float* __restrict__ J,
                                                   const float* __restrict__ K,
                                                   float* __restrict__ F) {
  int i = blockIdx.x * 256 + threadIdx.x;
  if (i < NBF2) F[i] = fmaf(2.0f, J[i], H[i]) - K[i];
}

// ---------- 96x96x96 fp32 GEMM on the matrix engine ----------
// One wave per 16x16 tile; 24 chained v_wmma_f32_16x16x4_f32 per tile.
// ISA layouts: A 16x4: lane l -> m=l&15, K pair base (l>>4)*2 in 2 VGPRs.
//              B 4x16 / C,D 16x16: row-striped across lanes; VGPR v -> rows v, v+8.
__global__ __launch_bounds__(32) void gemm96_wmma(const float* __restrict__ A,
                                                  const float* __restrict__ B,
                                                  float* __restrict__ C) {
  const int lane = threadIdx.x;
  const int tm = (blockIdx.x / 6) * 16;
  const int tn = (blockIdx.x % 6) * 16;
  const int mh = lane & 15;
  const int kb = (lane >> 4) * 2;
  v8f acc = {};
  for (int k0 = 0; k0 < NBF; k0 += 4) {
    v2f av, bv;
    av.x = A[(tm + mh) * NBF + k0 + kb];
    av.y = A[(tm + mh) * NBF + k0 + kb + 1];
    bv.x = B[(k0 + kb) * NBF + tn + mh];
    bv.y = B[(k0 + kb + 1) * NBF + tn + mh];
    acc = __builtin_amdgcn_wmma_f32_16x16x4_f32(false, av, false, bv,
                                                (short)0, acc, false, false);
  }
  const int mo = (lane >> 4) * 8;
  #pragma unroll
  for (int v = 0; v < 8; ++v)
    C[(tm + mo + v) * NBF + tn + mh] = acc[v];
}

// ---------- M = I - Fp/||Fp||_F  (lowest eig of Fp -> dominant eig of M) ----------
__global__ __launch_bounds__(256) void prep_kernel(const float* __restrict__ Fp,
                                                   float* __restrict__ M) {
  __shared__ float red[256];
  __shared__ float sinv;
  int t = threadIdx.x;
  float acc = 0.0f;
  for (int i = t; i < NBF2; i += 256) { float x = Fp[i]; acc = fmaf(x, x, acc); }
  red[t] = acc; __syncthreads();
  for (int off = 128; off; off >>= 1) { if (t < off) red[t] += red[t + off]; __syncthreads(); }
  if (t == 0) sinv = 1.0f / sqrtf(fmaxf(red[0], 1e-30f));
  __syncthreads();
  float si = sinv;
  for (int i = t; i < NBF2; i += 256) {
    int r = i / NBF, c = i - r * NBF;
    M[i] = (r == c ? 1.0f : 0.0f) - Fp[i] * si;
  }
}

// ---------- dst = src / max|src|  (keeps repeated squaring in range) ----------
__global__ __launch_bounds__(256) void scale_kernel(const float* __restrict__ src,
                                                    float* __restrict__ dst) {
  __shared__ float red[256];
  __shared__ float inv;
  int t = threadIdx.x;
  float m = 0.0f;
  for (int i = t; i < NBF2; i += 256) m = fmaxf(m, fabsf(src[i]));
  red[t] = m; __syncthreads();
  for (int off = 128; off; off >>= 1) { if (t < off) red[t] = fmaxf(red[t], red[t + off]); __syncthreads(); }
  if (t == 0) inv = 1.0f / fmaxf(red[0], 1e-30f);
  __syncthreads();
  float s = inv;
  for (int i = t; i < NBF2; i += 256) dst[i] = src[i] * s;
}

// ---------- v = normalize(M @ 1); c = A v; D = c c^T; E = <F+H, D> + Enuc ----------
__global__ __launch_bounds__(NBF) void eig_kernel(const float* __restrict__ M,
                                                  const float* __restrict__ A,
                                                  const float* __restrict__ F,
                                                  const float* __restrict__ H,
                                                  const float* __restrict__ Enuc,
                                                  float* __restrict__ Dm,
                                                  float* __restrict__ out) {
  __shared__ float v[NBF], c[NBF], red[NBF];
  int t = threadIdx.x;
  float acc = 0.0f;
  for (int j = 0; j < NBF; ++j) acc += M[t * NBF + j];   // M^(2^k) columns ~ eigvec
  float nrm = reduce96(red, t, acc * acc);
  v[t] = acc * rsqrtf(fmaxf(nrm, 1e-30f));
  __syncthreads();
  float ct = 0.0f;
  for (int j = 0; j < NBF; ++j) ct = fmaf(A[t * NBF + j], v[j], ct);
  c[t] = ct;
  __syncthreads();
  float ep = 0.0f;
  for (int j = 0; j < NBF; ++j) {
    float d = ct * c[j];
    Dm[t * NBF + j] = d;
    ep = fmaf(F[t * NBF + j] + H[t * NBF + j], d, ep);
  }
  float e = reduce96(red, t, ep);
  if (t == 0) out[0] = e + Enuc[0];
}

extern "C" void kernel_launch(void* const* d_in, const int* in_sizes, int n_in,
                              void* d_out, int out_size, void* d_ws, size_t ws_size,
                              hipStream_t stream) {
  (void)in_sizes; (void)n_in; (void)out_size; (void)ws_size;
  // setup_inputs order: D, H, A, G, Enuc (all fp32)
  const float* H    = (const float*)d_in[1];
  const float* A    = (const float*)d_in[2];
  const float* G    = (const float*)d_in[3];
  const float* Enuc = (const float*)d_in[4];
  float* out = (float*)d_out;

  float* ws = (float*)d_ws;
  float* Dm = ws + 0 * NBF2;
  float* J  = ws + 1 * NBF2;   // J,K contiguous -> one zero pass
  float* K  = ws + 2 * NBF2;
  float* F  = ws + 3 * NBF2;
  float* T  = ws + 4 * NBF2;
  float* Fp = ws + 5 * NBF2;
  float* M  = ws + 6 * NBF2;
  float* M2 = ws + 7 * NBF2;

  zero_kernel<<<(NBF2 + 255) / 256, 256, 0, stream>>>(Dm, NBF2);  // D starts at 0

  for (int it = 0; it < NITER; ++it) {
    zero_kernel<<<(2 * NBF2 + 255) / 256, 256, 0, stream>>>(J, 2 * NBF2);
    jk_kernel<<<NBF2, NBF, 0, stream>>>(G, Dm, J, K);
    fock_kernel<<<(NBF2 + 255) / 256, 256, 0, stream>>>(H, J, K, F);
    gemm96_wmma<<<36, 32, 0, stream>>>(A, F, T);    // T  = A F
    gemm96_wmma<<<36, 32, 0, stream>>>(T, A, Fp);   // Fp = T A
    prep_kernel<<<1, 256, 0, stream>>>(Fp, M);
    for (int s = 0; s < SQ_ITERS; ++s) {            // M <- (M M)/max|.|
      gemm96_wmma<<<36, 32, 0, stream>>>(M, M, M2);
      scale_kernel<<<1, 256, 0, stream>>>(M2, M);
    }
    eig_kernel<<<1, NBF, 0, stream>>>(M, A, F, H, Enuc, Dm, out);
  }
}